// IndRNNCell_onlyrecurrent_86517821212555
// MI455X (gfx1250) — compile-verified
//
#include <hip/hip_runtime.h>
#include <hip/hip_bf16.h>

// out = relu(input + hx * w[None,:]) , fp32, B=16384 H=2048.
// HBM-bound streaming kernel. Fast path: 2D grid (cols x row-groups),
// R=4 batch rows per thread sharing one cached weight float4 -> no integer
// division, 8 NT b128 loads + 4 NT b128 stores in flight per thread.

typedef float f32x4 __attribute__((ext_vector_type(4)));

__device__ __forceinline__ f32x4 fma_relu4(f32x4 a, f32x4 b, f32x4 wv) {
    f32x4 r;
#pragma unroll
    for (int k = 0; k < 4; ++k) {
        float v = __builtin_fmaf(b[k], wv[k], a[k]);
        r[k] = v > 0.0f ? v : 0.0f;
    }
    return r;
}

// Fast path: h4 % blockDim.x == 0 and rows % R == 0.
// grid = (h4/256, rows/R); each thread owns one float4 column x R rows.
template <int R>
__global__ __launch_bounds__(256) void indrnn_relu_tiled_kernel(
    const f32x4* __restrict__ in,
    const f32x4* __restrict__ hx,
    const f32x4* __restrict__ w,
    f32x4* __restrict__ out,
    unsigned int h4)                 // H/4 (row length in float4s)
{
    const unsigned int col4 = blockIdx.x * blockDim.x + threadIdx.x; // < h4
    const unsigned int row0 = blockIdx.y * R;
    unsigned long long base = (unsigned long long)row0 * h4 + col4;

    // Broadcast weight: 8 KB hot set, regular-temporal load (cache-resident),
    // loaded ONCE and reused across R rows.
    const f32x4 wv = w[col4];

    f32x4 a[R], b[R];
#pragma unroll
    for (int r = 0; r < R; ++r) {    // issue all streaming loads back-to-back
        a[r] = __builtin_nontemporal_load(&in[base + (unsigned long long)r * h4]);
        b[r] = __builtin_nontemporal_load(&hx[base + (unsigned long long)r * h4]);
    }
#pragma unroll
    for (int r = 0; r < R; ++r) {
        f32x4 res = fma_relu4(a[r], b[r], wv);
        __builtin_nontemporal_store(res, &out[base + (unsigned long long)r * h4]);
    }
}

// Generic fallback (any shape): one float4 per thread, modulo weight index.
__global__ __launch_bounds__(256) void indrnn_relu_generic_kernel(
    const f32x4* __restrict__ in,
    const f32x4* __restrict__ hx,
    const f32x4* __restrict__ w,
    f32x4* __restrict__ out,
    unsigned int n4,
    unsigned int h4)
{
    unsigned int i = blockIdx.x * blockDim.x + threadIdx.x;
    if (i >= n4) return;
    f32x4 a  = __builtin_nontemporal_load(&in[i]);
    f32x4 b  = __builtin_nontemporal_load(&hx[i]);
    f32x4 wv = w[i % h4];
    __builtin_nontemporal_store(fma_relu4(a, b, wv), &out[i]);
}

extern "C" void kernel_launch(void* const* d_in, const int* in_sizes, int n_in,
                              void* d_out, int out_size, void* d_ws, size_t ws_size,
                              hipStream_t stream) {
    const f32x4* in = (const f32x4*)d_in[0];   // input  (B*H fp32)
    const f32x4* hx = (const f32x4*)d_in[1];   // hx     (B*H fp32)
    const f32x4* w  = (const f32x4*)d_in[2];   // weight (H fp32)
    f32x4* out      = (f32x4*)d_out;

    const unsigned int n    = (unsigned int)in_sizes[0];   // B*H
    const unsigned int h    = (unsigned int)in_sizes[2];   // H
    const unsigned int n4   = n / 4;
    const unsigned int h4   = h / 4;                       // 512
    const unsigned int rows = n / h;                       // B = 16384

    const unsigned int block = 256;                        // 8 wave32s
    constexpr int R = 4;

    if ((h4 % block == 0) && (rows % R == 0)) {
        dim3 grid(h4 / block, rows / R);                   // (2, 4096)
        indrnn_relu_tiled_kernel<R><<<grid, block, 0, stream>>>(in, hx, w, out, h4);
    } else {
        dim3 grid((n4 + block - 1) / block);
        indrnn_relu_generic_kernel<<<grid, block, 0, stream>>>(in, hx, w, out, n4, h4);
    }
}